// PAM_Module_10350871183714
// MI455X (gfx1250) — compile-verified
//
#include <hip/hip_runtime.h>
#include <hip/hip_bf16.h>
#include <math.h>

typedef float v2f __attribute__((ext_vector_type(2)));
typedef float v4f __attribute__((ext_vector_type(4)));
typedef float v8f __attribute__((ext_vector_type(8)));

#define B_  8
#define C_  512
#define Cq_ 64
#define H_  128
#define W_  128
#define S_  256   // H+W

// ---------------------------------------------------------------------------
// Kernel 1: xk[b,c,s] = kronecker(x).  s<128: mean over w; s>=128: mean over h.
// One block (8 waves) per (b,c) plane. Each wave owns 16 rows:
//   - v4f (b128) nontemporal loads, fully coalesced
//   - row sums: in-register + 5-step wave32 shfl_xor reduction
//   - col sums: per-lane v4f register accumulators, combined via 4KB LDS
// ---------------------------------------------------------------------------
__global__ __launch_bounds__(256) void kron_reduce(const float* __restrict__ x,
                                                   float* __restrict__ xk) {
    __shared__ float part[8][128];          // per-wave column partial sums
    const int bc   = blockIdx.x;
    const int tid  = threadIdx.x;
    const int wv   = tid >> 5;
    const int lane = tid & 31;
    const v4f* p = (const v4f*)(x + (size_t)bc * (H_ * W_)); // 128 rows x 32 v4f

    v4f cacc = (v4f)(0.f);
    #pragma unroll 4
    for (int r = 0; r < 16; ++r) {
        const int row = wv * 16 + r;
        v4f v = __builtin_nontemporal_load(&p[row * 32 + lane]);
        cacc += v;
        float rs = (v.x + v.y) + (v.z + v.w);
        #pragma unroll
        for (int off = 16; off > 0; off >>= 1)
            rs += __shfl_xor(rs, off, 32);
        if (lane == 0)
            xk[(size_t)bc * S_ + row] = rs * (1.f / 128.f);
    }
    part[wv][lane * 4 + 0] = cacc.x;
    part[wv][lane * 4 + 1] = cacc.y;
    part[wv][lane * 4 + 2] = cacc.z;
    part[wv][lane * 4 + 3] = cacc.w;
    __syncthreads();
    if (tid < 128) {
        float s = 0.f;
        #pragma unroll
        for (int w8 = 0; w8 < 8; ++w8) s += part[w8][tid];
        xk[(size_t)bc * S_ + 128 + tid] = s * (1.f / 128.f);
    }
}

// ---------------------------------------------------------------------------
// Kernel 2: projection GEMM  out[b,m,n] = sum_k Wm[m,k]*xk[b,k,n] + bias[m]
// One wave per 16x16 tile, fp32 WMMA, K stepped by 4.
// ---------------------------------------------------------------------------
__global__ __launch_bounds__(256) void proj_gemm(const float* __restrict__ Wm,
                                                 const float* __restrict__ bias,
                                                 const float* __restrict__ xk,
                                                 float* __restrict__ out, int Mtiles) {
    const int wave = threadIdx.x >> 5;
    const int lane = threadIdx.x & 31;
    const int tile = blockIdx.x * 8 + wave;
    const int nt = tile & 15;
    const int t2 = tile >> 4;
    const int mt = t2 % Mtiles;
    const int b  = t2 / Mtiles;
    const int half = lane >> 4;
    const int l15  = lane & 15;
    const int n  = nt * 16 + l15;
    const int m  = mt * 16 + l15;
    const int kb = half * 2;
    const float* Wrow = Wm + (size_t)m * C_;
    const float* xb   = xk + (size_t)b * C_ * S_;
    v8f c;
    #pragma unroll
    for (int i = 0; i < 8; ++i) c[i] = bias[mt * 16 + i + 8 * half];
    for (int k0 = 0; k0 < C_; k0 += 4) {
        v2f a, bb;
        a.x  = Wrow[k0 + kb];
        a.y  = Wrow[k0 + kb + 1];
        bb.x = xb[(size_t)(k0 + kb) * S_ + n];
        bb.y = xb[(size_t)(k0 + kb + 1) * S_ + n];
        c = __builtin_amdgcn_wmma_f32_16x16x4_f32(false, a, false, bb,
                                                  (short)0, c, false, false);
    }
    const int M = Mtiles * 16;
    float* ob = out + (size_t)b * M * S_;
    #pragma unroll
    for (int i = 0; i < 8; ++i) {
        int mr = mt * 16 + i + 8 * half;
        ob[(size_t)mr * S_ + n] = c[i];
    }
}

// ---------------------------------------------------------------------------
// Kernel 3: energy[b,s,t] = sum_c q[b,c,s]*k[b,c,t]   (K = 64)
// ---------------------------------------------------------------------------
__global__ __launch_bounds__(256) void energy_gemm(const float* __restrict__ q,
                                                   const float* __restrict__ k,
                                                   float* __restrict__ energy) {
    const int wave = threadIdx.x >> 5;
    const int lane = threadIdx.x & 31;
    const int tile = blockIdx.x * 8 + wave;
    const int tt = tile & 15;
    const int t2 = tile >> 4;
    const int st = t2 & 15;
    const int b  = t2 >> 4;
    const int half = lane >> 4;
    const int l15  = lane & 15;
    const int sidx = st * 16 + l15;
    const int tidx = tt * 16 + l15;
    const int kb   = half * 2;
    const float* qb = q + (size_t)b * Cq_ * S_;
    const float* kp = k + (size_t)b * Cq_ * S_;
    v8f c = {};
    for (int k0 = 0; k0 < Cq_; k0 += 4) {
        v2f a, bb;
        a.x  = qb[(size_t)(k0 + kb) * S_ + sidx];     // A = q^T: [s][c]
        a.y  = qb[(size_t)(k0 + kb + 1) * S_ + sidx];
        bb.x = kp[(size_t)(k0 + kb) * S_ + tidx];     // B = k: [c][t]
        bb.y = kp[(size_t)(k0 + kb + 1) * S_ + tidx];
        c = __builtin_amdgcn_wmma_f32_16x16x4_f32(false, a, false, bb,
                                                  (short)0, c, false, false);
    }
    float* eb = energy + (size_t)b * S_ * S_;
    #pragma unroll
    for (int i = 0; i < 8; ++i) {
        int sr = st * 16 + i + 8 * half;
        eb[(size_t)sr * S_ + tidx] = c[i];
    }
}

// ---------------------------------------------------------------------------
// Kernel 4: in-place softmax over last axis; one 256-thread block per row.
// ---------------------------------------------------------------------------
__global__ __launch_bounds__(256) void softmax_rows(float* __restrict__ att) {
    __shared__ float red[256];
    const int tid = threadIdx.x;
    float* p = att + (size_t)blockIdx.x * S_;
    float v = p[tid];
    red[tid] = v; __syncthreads();
    for (int off = 128; off > 0; off >>= 1) {
        if (tid < off) red[tid] = fmaxf(red[tid], red[tid + off]);
        __syncthreads();
    }
    float mx = red[0]; __syncthreads();
    float e = __expf(v - mx);
    red[tid] = e; __syncthreads();
    for (int off = 128; off > 0; off >>= 1) {
        if (tid < off) red[tid] += red[tid + off];
        __syncthreads();
    }
    p[tid] = e * (1.f / red[0]);
}

// ---------------------------------------------------------------------------
// Kernel 5: out[b,c,t] = sum_s v[b,c,s]*att[b,t,s]   (K = 256)
// ---------------------------------------------------------------------------
__global__ __launch_bounds__(256) void out_gemm(const float* __restrict__ v,
                                                const float* __restrict__ att,
                                                float* __restrict__ outct) {
    const int wave = threadIdx.x >> 5;
    const int lane = threadIdx.x & 31;
    const int tile = blockIdx.x * 8 + wave;
    const int nt = tile & 15;
    const int t2 = tile >> 4;
    const int mt = t2 & 31;
    const int b  = t2 >> 5;
    const int half = lane >> 4;
    const int l15  = lane & 15;
    const int m  = mt * 16 + l15;   // channel c
    const int n  = nt * 16 + l15;   // output t
    const int kb = half * 2;
    const float* vb = v   + (size_t)b * C_ * S_;
    const float* ab = att + (size_t)b * S_ * S_;
    v8f c = {};
    for (int k0 = 0; k0 < S_; k0 += 4) {
        v2f a, bb;
        a.x  = vb[(size_t)m * S_ + k0 + kb];          // A = v: [c][s]
        a.y  = vb[(size_t)m * S_ + k0 + kb + 1];
        bb.x = ab[(size_t)n * S_ + k0 + kb];          // B = att^T: [s][t]
        bb.y = ab[(size_t)n * S_ + k0 + kb + 1];
        c = __builtin_amdgcn_wmma_f32_16x16x4_f32(false, a, false, bb,
                                                  (short)0, c, false, false);
    }
    float* ob = outct + (size_t)b * C_ * S_;
    #pragma unroll
    for (int i = 0; i < 8; ++i) {
        int mr = mt * 16 + i + 8 * half;
        ob[(size_t)mr * S_ + n] = c[i];
    }
}

// ---------------------------------------------------------------------------
// Kernel 6: inverse_kronecker + gamma*out + x, fused with the min-reductions.
// One block per (b,c) plane; v4f nontemporal streaming for x and y.
// ---------------------------------------------------------------------------
__global__ __launch_bounds__(256) void finalize(const float* __restrict__ outct,
                                                const float* __restrict__ x,
                                                const float* __restrict__ gamma,
                                                float* __restrict__ y) {
    __shared__ float sp[256];   // preserved h_part / w_part values
    __shared__ float sr[256];   // reduction scratch
    const int bc  = blockIdx.x;
    const int tid = threadIdx.x;
    float vv = outct[(size_t)bc * S_ + tid];
    sp[tid] = vv; sr[tid] = vv;
    __syncthreads();
    const int grp  = tid >> 7;        // 0 = h half, 1 = w half
    const int idx  = tid & 127;
    const int base = grp << 7;
    for (int off = 64; off > 0; off >>= 1) {
        if (idx < off) sr[base + idx] = fminf(sr[base + idx], sr[base + idx + off]);
        __syncthreads();
    }
    const float sub = sr[0] + sr[128];     // h_min + w_min
    const float g   = gamma[0];
    const v4f* xp = (const v4f*)(x + (size_t)bc * (H_ * W_));
    v4f*       yp = (v4f*)(y + (size_t)bc * (H_ * W_));
    // plane = 4096 v4f's; 16 per thread
    for (int i = tid; i < (H_ * W_) / 4; i += 256) {
        const int h = i >> 5;             // (i*4) >> 7
        const int w = (i & 31) * 4;       // (i*4) & 127
        v4f xv = __builtin_nontemporal_load(&xp[i]);
        const float hp = sp[h] - sub;
        v4f o;
        o.x = g * (sp[128 + w + 0] + hp) + xv.x;
        o.y = g * (sp[128 + w + 1] + hp) + xv.y;
        o.z = g * (sp[128 + w + 2] + hp) + xv.z;
        o.w = g * (sp[128 + w + 3] + hp) + xv.w;
        __builtin_nontemporal_store(o, &yp[i]);
    }
}

// ---------------------------------------------------------------------------
extern "C" void kernel_launch(void* const* d_in, const int* in_sizes, int n_in,
                              void* d_out, int out_size, void* d_ws, size_t ws_size,
                              hipStream_t stream) {
    const float* x     = (const float*)d_in[0];
    const float* Wq    = (const float*)d_in[1];
    const float* bq    = (const float*)d_in[2];
    const float* Wk    = (const float*)d_in[3];
    const float* bk    = (const float*)d_in[4];
    const float* Wv    = (const float*)d_in[5];
    const float* bv    = (const float*)d_in[6];
    const float* gamma = (const float*)d_in[7];
    float* y  = (float*)d_out;
    float* ws = (float*)d_ws;

    float* xk = ws;                             // 8*512*256  = 1,048,576 f
    float* q  = xk + (size_t)B_ * C_  * S_;     // 8*64*256   =   131,072 f
    float* k  = q  + (size_t)B_ * Cq_ * S_;     //              131,072 f
    float* v  = k  + (size_t)B_ * Cq_ * S_;     // 8*512*256  = 1,048,576 f
    float* en = v  + (size_t)B_ * C_  * S_;     // 8*256*256  =   524,288 f
    float* oc = en + (size_t)B_ * S_  * S_;     // 8*512*256  = 1,048,576 f
                                                // total ~15 MB of workspace

    kron_reduce<<<B_ * C_, 256, 0, stream>>>(x, xk);

    proj_gemm<<<(B_ * (Cq_ / 16) * (S_ / 16)) / 8, 256, 0, stream>>>(Wq, bq, xk, q, Cq_ / 16);
    proj_gemm<<<(B_ * (Cq_ / 16) * (S_ / 16)) / 8, 256, 0, stream>>>(Wk, bk, xk, k, Cq_ / 16);
    proj_gemm<<<(B_ * (C_  / 16) * (S_ / 16)) / 8, 256, 0, stream>>>(Wv, bv, xk, v, C_ / 16);

    energy_gemm<<<(B_ * (S_ / 16) * (S_ / 16)) / 8, 256, 0, stream>>>(q, k, en);
    softmax_rows<<<B_ * S_, 256, 0, stream>>>(en);
    out_gemm<<<(B_ * (C_ / 16) * (S_ / 16)) / 8, 256, 0, stream>>>(v, en, oc);

    finalize<<<B_ * C_, 256, 0, stream>>>(oc, x, gamma, y);
}